// SparseSelfAttention_82617990906534
// MI455X (gfx1250) — compile-verified
//
#include <hip/hip_runtime.h>

#define EMBED 128
#define HEADS 8

typedef __attribute__((ext_vector_type(2))) float v2f;
typedef __attribute__((ext_vector_type(8))) float v8f;

// Monotonic float <-> uint mapping so atomicMax(uint) implements float max.
__device__ __forceinline__ unsigned enc_f32(float f) {
    unsigned u = __float_as_uint(f);
    return (u & 0x80000000u) ? ~u : (u | 0x80000000u);
}
__device__ __forceinline__ float dec_f32(unsigned u) {
    return (u & 0x80000000u) ? __uint_as_float(u & 0x7FFFFFFFu)
                             : __uint_as_float(~u);
}

// ---------------------------------------------------------------------------
// Out[N,128] = A[N,128] @ W[128,128]^T + bias, via V_WMMA_F32_16X16X4_F32.
// One block = 8 waves; wave w computes the 16x16 output tile at column w*16.
// A tile (16x4 f32): lane L holds A[row0 + (L&15)][4k + 2*(L>>4) + {0,1}]
// B tile (4x16 f32): lane L holds W[col0 + (L&15)][4k + 2*(L>>4) + {0,1}]
// D tile: VGPR r of lane L = D[r + 8*(L>>4)][L&15]
// ---------------------------------------------------------------------------
__global__ __launch_bounds__(256) void gemm_xwT_bias(
    const float* __restrict__ A,    // [N,128]
    const float* __restrict__ W,    // [128,128] row-major
    const float* __restrict__ bias, // [128]
    float* __restrict__ Out,        // [N,128]
    int N)
{
    const int lane  = threadIdx.x & 31;
    const int wave  = threadIdx.x >> 5;     // 0..7 -> column tile
    const int row0  = blockIdx.x * 16;
    const int col0  = wave * 16;

    const int mj    = lane & 15;            // M index for A, N index for B/D
    const int khalf = (lane >> 4) << 1;     // 0 or 2

    int arow = row0 + mj;
    if (arow >= N) arow = N - 1;            // clamp for safe (dummy) loads
    const float* aPtr = A + (size_t)arow * EMBED + khalf;
    const float* bPtr = W + (size_t)(col0 + mj) * EMBED + khalf;

    v8f c = {};
#pragma unroll
    for (int k = 0; k < EMBED; k += 4) {
        v2f a = *(const v2f*)(aPtr + k);
        v2f b = *(const v2f*)(bPtr + k);
        // 8 args: (neg_a, A, neg_b, B, c_mod, C, reuse_a, reuse_b)
        c = __builtin_amdgcn_wmma_f32_16x16x4_f32(
                false, a, false, b, (short)0, c, false, false);
    }

    const float bval  = bias[col0 + mj];
    const int   mbase = row0 + ((lane >> 4) << 3);
#pragma unroll
    for (int r = 0; r < 8; ++r) {
        int row = mbase + r;
        if (row < N) Out[(size_t)row * EMBED + col0 + mj] = c[r] + bval;
    }
}

// ---------------------------------------------------------------------------
// One wave per edge. Lane l covers head h = l>>2, sub-chunk l&3 (float4).
// logits[e,h] = 0.25 * dot(Q[dst,h*16:+16], K[src,h*16:+16]) + bias[e,h]
// Also maintains per-(dst,h) running max via uint atomicMax.
// ---------------------------------------------------------------------------
__global__ __launch_bounds__(256) void edge_logits_kernel(
    const int*   __restrict__ edge_index, // [2,E]
    const float* __restrict__ attn_bias,  // [E,8]
    const float* __restrict__ Q,
    const float* __restrict__ Kp,
    float*       __restrict__ logits,     // [E,8]
    unsigned*    __restrict__ mEnc,       // [N,8]
    int E)
{
    const int lane = threadIdx.x & 31;
    const int e    = blockIdx.x * 8 + (threadIdx.x >> 5);
    if (e >= E) return;

    const int dst = edge_index[e];
    const int src = edge_index[E + e];

    const float4 q = *(const float4*)(Q  + (size_t)dst * EMBED + lane * 4);
    const float4 k = *(const float4*)(Kp + (size_t)src * EMBED + lane * 4);
    float p = q.x * k.x + q.y * k.y + q.z * k.z + q.w * k.w;
    p += __shfl_xor(p, 1, 32);
    p += __shfl_xor(p, 2, 32);

    if ((lane & 3) == 0) {
        const int h = lane >> 2;
        float lg = 0.25f * p + attn_bias[(size_t)e * HEADS + h]; // 1/sqrt(16)
        logits[(size_t)e * HEADS + h] = lg;
        atomicMax(&mEnc[(size_t)dst * HEADS + h], enc_f32(lg));
    }
}

// ---------------------------------------------------------------------------
// One thread per (edge, head): z[dst,h] += exp(logit - m[dst,h])
// ---------------------------------------------------------------------------
__global__ __launch_bounds__(256) void edge_expsum_kernel(
    const int*      __restrict__ edge_index,
    const float*    __restrict__ logits,
    const unsigned* __restrict__ mEnc,
    float*          __restrict__ z,
    int E)
{
    const int idx = blockIdx.x * blockDim.x + threadIdx.x; // e*8 + h
    if (idx >= E * HEADS) return;
    const int e = idx >> 3, h = idx & 7;
    const int dst = edge_index[e];
    const float m = dec_f32(mEnc[(size_t)dst * HEADS + h]);
    const float ev = __expf(logits[idx] - m);
    unsafeAtomicAdd(&z[(size_t)dst * HEADS + h], ev);
}

// ---------------------------------------------------------------------------
// One wave per edge: agg[dst] += softmax_score(e,h) * V[src] (float4 per lane)
// ---------------------------------------------------------------------------
__global__ __launch_bounds__(256) void edge_agg_kernel(
    const int*      __restrict__ edge_index,
    const float*    __restrict__ logits,
    const unsigned* __restrict__ mEnc,
    const float*    __restrict__ z,
    const float*    __restrict__ V,
    float*          __restrict__ agg,
    int E)
{
    const int lane = threadIdx.x & 31;
    const int e    = blockIdx.x * 8 + (threadIdx.x >> 5);
    if (e >= E) return;

    const int dst = edge_index[e];
    const int src = edge_index[E + e];
    const int h   = lane >> 2;

    const float m  = dec_f32(mEnc[(size_t)dst * HEADS + h]);
    const float s  = __expf(logits[(size_t)e * HEADS + h] - m)
                     / z[(size_t)dst * HEADS + h];
    const float4 v = *(const float4*)(V + (size_t)src * EMBED + lane * 4);

    float* a = agg + (size_t)dst * EMBED + lane * 4;
    unsafeAtomicAdd(a + 0, s * v.x);
    unsafeAtomicAdd(a + 1, s * v.y);
    unsafeAtomicAdd(a + 2, s * v.z);
    unsafeAtomicAdd(a + 3, s * v.w);
}

// ---------------------------------------------------------------------------
extern "C" void kernel_launch(void* const* d_in, const int* in_sizes, int n_in,
                              void* d_out, int out_size, void* d_ws, size_t ws_size,
                              hipStream_t stream) {
    (void)n_in; (void)out_size; (void)ws_size;

    const float* x  = (const float*)d_in[0];
    const int*   ei = (const int*)  d_in[1];
    const float* ab = (const float*)d_in[2];
    const float* Wq = (const float*)d_in[3];
    const float* bq = (const float*)d_in[4];
    const float* Wk = (const float*)d_in[5];
    const float* bk = (const float*)d_in[6];
    const float* Wv = (const float*)d_in[7];
    const float* bv = (const float*)d_in[8];
    const float* Wo = (const float*)d_in[9];
    const float* bo = (const float*)d_in[10];

    const int N = in_sizes[0] / EMBED;   // 25000
    const int E = in_sizes[1] / 2;       // 400000

    float* out    = (float*)d_out;                 // [N,128]
    float* logits = out + (size_t)N * EMBED;       // [E,8]

    // Workspace: Q | K | V | mEnc | z | agg
    float*    ws    = (float*)d_ws;
    const size_t n128 = (size_t)N * EMBED;
    const size_t nh   = (size_t)N * HEADS;
    float*    Q     = ws;
    float*    Kp    = ws + n128;
    float*    V     = ws + 2 * n128;
    unsigned* mEnc  = (unsigned*)(ws + 3 * n128);
    float*    z     = ws + 3 * n128 + nh;
    float*    agg   = ws + 3 * n128 + 2 * nh;

    // Zero m (encoded-uint 0 == "-inf"), z and agg in one shot.
    hipMemsetAsync(mEnc, 0, (2 * nh + n128) * sizeof(float), stream);

    const int rowTiles = (N + 15) / 16;
    gemm_xwT_bias<<<rowTiles, 256, 0, stream>>>(x, Wq, bq, Q,  N);
    gemm_xwT_bias<<<rowTiles, 256, 0, stream>>>(x, Wk, bk, Kp, N);
    gemm_xwT_bias<<<rowTiles, 256, 0, stream>>>(x, Wv, bv, V,  N);

    const int edgeBlocks = (E + 7) / 8;
    edge_logits_kernel<<<edgeBlocks, 256, 0, stream>>>(ei, ab, Q, Kp, logits, mEnc, E);

    const int ehThreads = E * HEADS;
    edge_expsum_kernel<<<(ehThreads + 255) / 256, 256, 0, stream>>>(ei, logits, mEnc, z, E);

    edge_agg_kernel<<<edgeBlocks, 256, 0, stream>>>(ei, logits, mEnc, z, V, agg, E);

    gemm_xwT_bias<<<rowTiles, 256, 0, stream>>>(agg, Wo, bo, out, N);
}